// GAT_28295244546247
// MI455X (gfx1250) — compile-verified
//
#include <hip/hip_runtime.h>
#include <hip/hip_bf16.h>
#include <math.h>

// ---------------------------------------------------------------------------
// GAT forward, algebraically collapsed (see analysis):
//   attn rows are i-independent -> out1 is one row broadcast; layer 2 attention
//   becomes uniform -> final = leaky_relu(out1_row @ W2^T + b2) broadcast.
// Only real GEMM (e1 = x @ W1^T + b1) runs on fp32 WMMA (16x16x4).
// ---------------------------------------------------------------------------

typedef float v2f __attribute__((ext_vector_type(2)));
typedef float v8f __attribute__((ext_vector_type(8)));

#define N_NODES 4096
#define F_HID   64
#define F_OUT   32
#define N_HEADS 8
#define NEG_SLOPE 0.2f

// K1: e1 = x @ W1^T + b1   via V_WMMA_F32_16X16X4_F32
// grid: 256 blocks (16-row strips), 128 threads = 4 waves, wave w -> N-tile w.
__global__ __launch_bounds__(128) void gat_gemm1_wmma(
    const float* __restrict__ x, const float* __restrict__ W1,
    const float* __restrict__ b1, float* __restrict__ e1) {
  const int lane  = threadIdx.x & 31;
  const int wave  = threadIdx.x >> 5;
  const int j0    = blockIdx.x << 4;          // row strip base (M)
  const int m     = lane & 15;
  const int khalf = (lane >> 4) << 1;         // lanes 0-15 -> K{0,1}, 16-31 -> K{2,3}
  const int n     = (wave << 4) + m;          // output feature column (N)

  const float* xrow = x  + (j0 + m) * F_HID + khalf; // A: M=j0+m
  const float* wrow = W1 + n * F_HID + khalf;        // B[k,n] = W1[n,k]

  v8f c = {};
#pragma unroll
  for (int k = 0; k < F_HID; k += 4) {
    v2f a = *(const v2f*)(xrow + k);   // A 16x4 f32 layout (ISA 7.12.2)
    v2f b = *(const v2f*)(wrow + k);   // B 4x16 f32, mirrored layout
    c = __builtin_amdgcn_wmma_f32_16x16x4_f32(
        /*neg_a=*/false, a, /*neg_b=*/false, b,
        /*c_mod=*/(short)0, c, /*reuse_a=*/false, /*reuse_b=*/false);
  }

  // C/D 16x16 f32: VGPR r -> (M=r, N=lane) lanes 0-15 ; (M=r+8, N=lane-16)
  const int  rbase = j0 + ((lane >> 4) << 3);
  const float bias = b1[n];
#pragma unroll
  for (int r = 0; r < 8; ++r)
    e1[(rbase + r) * F_HID + n] = c[r] + bias;
}

// K2: single workgroup: scores -> per-head softmax over j -> head-averaged
// weights -> weighted row-sum -> leaky_relu -> elu -> W2 GEMV -> leaky_relu.
__global__ __launch_bounds__(1024) void gat_reduce(
    const float* __restrict__ e1, const float* __restrict__ a1w,
    const float* __restrict__ W2, const float* __restrict__ b2,
    float* __restrict__ s, float* __restrict__ wb, float* __restrict__ frow) {
  __shared__ float red[1024];
  __shared__ float hmax[N_HEADS];
  __shared__ float hden[N_HEADS];
  __shared__ float aw[N_HEADS][F_HID];
  __shared__ float acc[F_HID * 16];
  __shared__ float act[F_HID];

  const int tid = threadIdx.x;

  // stage the "j" half of the per-head attention weights: a1w[h, F_HID + f]
  if (tid < N_HEADS * F_HID)
    aw[tid >> 6][tid & 63] = a1w[(tid >> 6) * (2 * F_HID) + F_HID + (tid & 63)];
  __syncthreads();

  // Phase 1: s[j,h] = e1[j,:] . aw[h,:]  and per-thread max per head
  float lmax[N_HEADS];
#pragma unroll
  for (int h = 0; h < N_HEADS; ++h) lmax[h] = -3.402823466e38f;

  for (int j = tid; j < N_NODES; j += 1024) {
    const float* row = e1 + j * F_HID;
    float sc[N_HEADS] = {};
    for (int f = 0; f < F_HID; ++f) {
      const float v = row[f];
#pragma unroll
      for (int h = 0; h < N_HEADS; ++h) sc[h] = fmaf(v, aw[h][f], sc[h]);
    }
#pragma unroll
    for (int h = 0; h < N_HEADS; ++h) {
      s[j * N_HEADS + h] = sc[h];
      lmax[h] = fmaxf(lmax[h], sc[h]);
    }
  }

  // per-head max reduction
  for (int h = 0; h < N_HEADS; ++h) {
    __syncthreads();
    red[tid] = lmax[h];
    __syncthreads();
    for (int off = 512; off > 0; off >>= 1) {
      if (tid < off) red[tid] = fmaxf(red[tid], red[tid + off]);
      __syncthreads();
    }
    if (tid == 0) hmax[h] = red[0];
  }
  __syncthreads();

  // Phase 2: per-head sum of exp
  float lsum[N_HEADS] = {};
  for (int j = tid; j < N_NODES; j += 1024) {
#pragma unroll
    for (int h = 0; h < N_HEADS; ++h)
      lsum[h] += expf(s[j * N_HEADS + h] - hmax[h]);
  }
  for (int h = 0; h < N_HEADS; ++h) {
    __syncthreads();
    red[tid] = lsum[h];
    __syncthreads();
    for (int off = 512; off > 0; off >>= 1) {
      if (tid < off) red[tid] += red[tid + off];
      __syncthreads();
    }
    if (tid == 0) hden[h] = red[0];
  }
  __syncthreads();

  // Phase 3a: head-averaged attention weights  wb[j] = (1/H) sum_h attn[h,j]
  float rden[N_HEADS];
#pragma unroll
  for (int h = 0; h < N_HEADS; ++h) rden[h] = 1.0f / hden[h];
  for (int j = tid; j < N_NODES; j += 1024) {
    float w = 0.f;
#pragma unroll
    for (int h = 0; h < N_HEADS; ++h)
      w += expf(s[j * N_HEADS + h] - hmax[h]) * rden[h];
    wb[j] = w * (1.0f / N_HEADS);
  }
  __syncthreads();

  // Phase 3b: out1_row[f] = sum_j wb[j] * e1[j,f]
  {
    const int f = tid & 63, chunk = tid >> 6;
    const int jb = chunk << 8;
    float p = 0.f;
    for (int jj = 0; jj < 256; ++jj)
      p = fmaf(wb[jb + jj], e1[(jb + jj) * F_HID + f], p);
    acc[f * 16 + chunk] = p;
  }
  __syncthreads();

  if (tid < F_HID) {
    float v = 0.f;
#pragma unroll
    for (int c2 = 0; c2 < 16; ++c2) v += acc[tid * 16 + c2];
    v = (v > 0.f) ? v : NEG_SLOPE * v;   // leaky_relu (inside MHA)
    v = (v > 0.f) ? v : expm1f(v);       // elu
    act[tid] = v;
  }
  __syncthreads();

  // Phase 4: e2_row = act @ W2^T + b2 ; final = leaky_relu(e2_row)
  // (layer-2 attention is uniform over identical rows -> identity on the row)
  if (tid < F_OUT) {
    float v = b2[tid];
    for (int f2 = 0; f2 < F_HID; ++f2)
      v = fmaf(W2[tid * F_HID + f2], act[f2], v);
    frow[tid] = (v > 0.f) ? v : NEG_SLOPE * v;
  }
}

// K3: broadcast the 32-float result row to all 4096 nodes (float4 stores).
__global__ __launch_bounds__(256) void gat_broadcast(
    const float* __restrict__ frow, float4* __restrict__ out) {
  __shared__ float r[F_OUT];
  if (threadIdx.x < F_OUT) r[threadIdx.x] = frow[threadIdx.x];
  __syncthreads();
  const int i = blockIdx.x * 256 + threadIdx.x;   // 4096*32/4 = 32768 float4s
  const int c = (i & 7) << 2;
  float4 v;
  v.x = r[c]; v.y = r[c + 1]; v.z = r[c + 2]; v.w = r[c + 3];
  out[i] = v;
}

extern "C" void kernel_launch(void* const* d_in, const int* in_sizes, int n_in,
                              void* d_out, int out_size, void* d_ws, size_t ws_size,
                              hipStream_t stream) {
  (void)in_sizes; (void)n_in; (void)out_size; (void)ws_size;
  const float* x    = (const float*)d_in[0];
  const float* W1   = (const float*)d_in[1];
  const float* b1   = (const float*)d_in[2];
  const float* a1_w = (const float*)d_in[3];
  // a1_b (d_in[4]) cancels in softmax; a2_w/a2_b (d_in[7..8]) have no effect
  const float* W2   = (const float*)d_in[5];
  const float* b2   = (const float*)d_in[6];

  float* ws   = (float*)d_ws;
  float* e1   = ws;                            // 4096*64
  float* s    = e1 + N_NODES * F_HID;          // 4096*8
  float* wb   = s + N_NODES * N_HEADS;         // 4096
  float* frow = wb + N_NODES;                  // 32

  gat_gemm1_wmma<<<N_NODES / 16, 128, 0, stream>>>(x, W1, b1, e1);
  gat_reduce<<<1, 1024, 0, stream>>>(e1, a1_w, W2, b2, s, wb, frow);
  gat_broadcast<<<(N_NODES * F_OUT / 4) / 256, 256, 0, stream>>>(
      frow, (float4*)d_out);
}